// Sampler_61675730370560
// MI455X (gfx1250) — compile-verified
//
#include <hip/hip_runtime.h>

// Problem constants (from the reference setup_inputs):
//   x:    [N=8, C=64, H=256, W=256] float32
//   grid: [N=8, Hg=256, Wg=256, 2] float32   (values in [-1,1], (x,y))
//   out:  [N=8, C=64, Hg=256, Wg=256] float32
#define NB   8
#define CC   64
#define HH   256
#define WW   256
#define HW   (HH * WW)      // 65536, also Hg*Wg
#define CGRP 8              // channel-loop unroll / prefetch group

// native clang vector type: accepted by __builtin_nontemporal_load and
// lowers to one global_load_b64 th:TH_LOAD_NT per lane.
typedef float v2f __attribute__((ext_vector_type(2)));

__global__ __launch_bounds__(256) void grid_sample_bilinear(
    const float* __restrict__ x,
    const float* __restrict__ grid,
    float* __restrict__ out)
{
    // One block = one output row (n, hg, 0..255). Lanes = consecutive wg ->
    // coalesced grid loads (b64/lane) and coalesced output stores (b32/lane).
    const int wg = threadIdx.x;          // 0..255
    const int nh = blockIdx.x;           // n*256 + hg, 0..2047
    const int hg = nh & (HH - 1);
    const int n  = nh >> 8;

    // ---- per-spatial-point setup (amortized over 64 channels) ----
    // grid is read exactly once -> non-temporal load, keep L2 for x.
    const int sIdx = (nh << 8) | wg;     // n*Hg*Wg + hg*Wg + wg
    const v2f g =
        __builtin_nontemporal_load(reinterpret_cast<const v2f*>(grid) + sIdx);

    // unnormalize, align_corners=False: ix = ((gx+1)*W - 1)/2 = gx*W/2 + (W-1)/2
    const float ix = fmaf(g.x, 0.5f * (float)WW, 0.5f * ((float)WW - 1.0f));
    const float iy = fmaf(g.y, 0.5f * (float)HH, 0.5f * ((float)HH - 1.0f));

    const float fx0 = floorf(ix);
    const float fy0 = floorf(iy);
    const float tx  = ix - fx0;
    const float ty  = iy - fy0;

    const int x0 = (int)fx0, y0 = (int)fy0;
    const int x1 = x0 + 1,   y1 = y0 + 1;

    // bilinear weights; zero out invalid corners (zero-padding), branch-free
    float wA = (1.0f - tx) * (1.0f - ty);   // (x0, y0)
    float wB = (1.0f - tx) * ty;            // (x0, y1)
    float wC = tx * (1.0f - ty);            // (x1, y0)
    float wD = tx * ty;                     // (x1, y1)

    const bool vx0 = (x0 >= 0) & (x0 < WW);
    const bool vx1 = (x1 >= 0) & (x1 < WW);
    const bool vy0 = (y0 >= 0) & (y0 < HH);
    const bool vy1 = (y1 >= 0) & (y1 < HH);
    wA = (vx0 & vy0) ? wA : 0.0f;
    wB = (vx0 & vy1) ? wB : 0.0f;
    wC = (vx1 & vy0) ? wC : 0.0f;
    wD = (vx1 & vy1) ? wD : 0.0f;

    // clamped gather indices -> all 4 loads unconditional, no divergence
    const int cx0 = min(max(x0, 0), WW - 1);
    const int cx1 = min(max(x1, 0), WW - 1);
    const int cy0 = min(max(y0, 0), HH - 1);
    const int cy1 = min(max(y1, 0), HH - 1);

    const int offA = cy0 * WW + cx0;
    const int offB = cy1 * WW + cx0;
    const int offC = cy0 * WW + cx1;
    const int offD = cy1 * WW + cx1;

    const float* __restrict__ xb = x + (size_t)n * CC * HW;          // image n
    float* __restrict__ ob = out + (size_t)n * CC * HW + hg * WW + wg;
    // out element for channel c lives at ob + c*HW

    // ---- channel loop: 4 gathers (RT, L2-resident) + 4 FMAs + 1 NT store ----
    for (int cg = 0; cg < CC; cg += CGRP) {
        // gfx1250 path: fire-and-forget global_prefetch of the next channel
        // group's corner cachelines (no LOADcnt cost) to warm the gather stream.
        if (cg + CGRP < CC) {
            const float* pf = xb + (cg + CGRP) * HW;
            __builtin_prefetch(pf + offA, 0, 0);
            __builtin_prefetch(pf + offB, 0, 0);
            __builtin_prefetch(pf + offC, 0, 0);
            __builtin_prefetch(pf + offD, 0, 0);
        }
#pragma unroll
        for (int u = 0; u < CGRP; ++u) {
            const int c = cg + u;
            const float* __restrict__ p = xb + c * HW;
            const float vA = p[offA];
            const float vB = p[offB];
            const float vC = p[offC];
            const float vD = p[offD];
            float r = vA * wA;
            r = fmaf(vB, wB, r);
            r = fmaf(vC, wC, r);
            r = fmaf(vD, wD, r);
            // Output is write-once/never-read: stream past L2 (TH_NT) so the
            // 134 MB of x stays resident in the 192 MB L2 for the gathers.
            __builtin_nontemporal_store(r, ob + c * HW);
        }
    }
}

extern "C" void kernel_launch(void* const* d_in, const int* in_sizes, int n_in,
                              void* d_out, int out_size, void* d_ws, size_t ws_size,
                              hipStream_t stream) {
    const float* x    = (const float*)d_in[0];   // [8,64,256,256] f32
    const float* grid = (const float*)d_in[1];   // [8,256,256,2]  f32
    float* out        = (float*)d_out;           // [8,64,256,256] f32

    const dim3 block(256, 1, 1);
    const dim3 gridDim(NB * HH, 1, 1);           // 2048 blocks: one per (n, hg) row
    hipLaunchKernelGGL(grid_sample_bilinear, gridDim, block, 0, stream,
                       x, grid, out);
}